// CML_87969520157217
// MI455X (gfx1250) — compile-verified
//
#include <hip/hip_runtime.h>

typedef __attribute__((ext_vector_type(2))) float v2f;
typedef __attribute__((ext_vector_type(8))) float v8f;

#define DIM          64
#define N_SCORE      256
#define N_ITEMS      500000
#define NTILES_I     (N_ITEMS / 16)   // 31250 item tiles of 16
#define WAVES_PER_BLK 8

// scores[m, n] = -(||u_m||^2 + ||i_n||^2 - 2 * <u_m, i_n>)
//             = 2*cross - u_sq - i_sq
__global__ __launch_bounds__(256) void cml_score_kernel(
    const int*   __restrict__ score_user_ids,
    const float* __restrict__ user_emb,
    const float* __restrict__ item_emb,
    float*       __restrict__ out)
{
    const int lane = threadIdx.x & 31;
    const int wave = threadIdx.x >> 5;
    const int r    = lane & 15;   // row within 16-tile (M for A, N for B / output col)
    const int half = lane >> 4;   // selects which K-pair this lane holds

    // ---- A tile: 16 users x 64 dims, held in registers in WMMA layout ----
    // V_WMMA_F32_16X16X4_F32 A layout: lane(0-15)->M, VGPR v holds K=2*half+v.
    // For K-step t (kbase = 4t): lane loads floats [4t + 2*half, 4t + 2*half + 1].
    const int utile = blockIdx.y;                       // 0..15
    const int uid   = score_user_ids[utile * 16 + r];
    const float* urow = user_emb + (size_t)uid * DIM;

    v2f a[16];
    float usq_part = 0.0f;
#pragma unroll
    for (int t = 0; t < 16; ++t) {
        a[t] = *(const v2f*)(urow + 4 * t + 2 * half);
        usq_part += a[t].x * a[t].x + a[t].y * a[t].y;
    }
    // Lanes L and L^16 hold complementary K-halves of the same row r.
    float usq = usq_part + __shfl_xor(usq_part, 16, 32);
    // D layout: VGPR j, this lane -> output row m = 8*half + j.  Pre-broadcast
    // the 8 loop-invariant u_sq values this lane will need (lane s<16 has usq of row s).
    float usq_m[8];
#pragma unroll
    for (int j = 0; j < 8; ++j)
        usq_m[j] = __shfl(usq, 8 * half + j, 32);

    float* out_base = out + (size_t)(utile * 16) * (size_t)N_ITEMS;

    for (int tile = blockIdx.x * WAVES_PER_BLK + wave; tile < NTILES_I;
         tile += gridDim.x * WAVES_PER_BLK) {
        const int icol0 = tile * 16;
        const float* irow = item_emb + (size_t)(icol0 + r) * DIM;

        v8f c = {};   // 16x16 f32 accumulator
        float isq_part = 0.0f;
#pragma unroll
        for (int t = 0; t < 16; ++t) {
            v2f b = *(const v2f*)(irow + 4 * t + 2 * half);
            isq_part += b.x * b.x + b.y * b.y;
            // D = A x B + C, exact f32 path: v_wmma_f32_16x16x4_f32
            c = __builtin_amdgcn_wmma_f32_16x16x4_f32(
                    /*neg_a=*/false, a[t], /*neg_b=*/false, b,
                    /*c_mod=*/(short)0, c, /*reuse_a=*/false, /*reuse_b=*/false);
        }
        float isq = isq_part + __shfl_xor(isq_part, 16, 32);

        // D layout: VGPR j, lane -> M = 8*half + j, N = r. Non-temporal stores:
        // 512MB of output must not evict the 128MB item table from L2.
#pragma unroll
        for (int j = 0; j < 8; ++j) {
            float s = 2.0f * c[j] - usq_m[j] - isq;
            __builtin_nontemporal_store(
                s, out_base + (size_t)(8 * half + j) * (size_t)N_ITEMS + icol0 + r);
        }
    }
}

extern "C" void kernel_launch(void* const* d_in, const int* in_sizes, int n_in,
                              void* d_out, int out_size, void* d_ws, size_t ws_size,
                              hipStream_t stream) {
    const int*   ids  = (const int*)d_in[0];    // score_user_ids [256]
    const float* uemb = (const float*)d_in[1];  // user_embeddings [100000,64]
    const float* iemb = (const float*)d_in[2];  // item_embeddings [500000,64]
    float* out = (float*)d_out;                 // [256, 500000]

    dim3 block(256, 1, 1);                      // 8 waves
    dim3 grid(128, N_SCORE / 16, 1);            // item-tile strided loop x 16 user tiles
    hipLaunchKernelGGL(cml_score_kernel, grid, block, 0, stream,
                       ids, uemb, iemb, out);
}